// CostVolumeLayer_70849780515366
// MI455X (gfx1250) — compile-verified
//
#include <hip/hip_runtime.h>

// ---------------------------------------------------------------------------
// Cost volume (search range 4) for MI455X / gfx1250 — full-f32 WMMA version.
// out[b,(di,dj),h,w] = 1/(C*81) * sum_c x[b,c,h,w] * warped_pad[b,c,h+di,w+dj]
//
// Roofline: ~341 MB min HBM traffic -> ~15 us @ 23.3 TB/s. The 5.4 GFLOP of
// channel dots needs ~370 TFLOPS to keep pace -> matrix pipes. Using
// v_wmma_f32_16x16x4_f32 keeps exact f32 math (matches reference) and makes
// staging a pure byte copy, so tiles stream global->LDS via the CDNA5 async
// path (ASYNCcnt) with no VGPR round-trip and no conversion VALU work.
// Per (b, h, 16-wide w strip, di): two 16x16 product tiles whose dj-band is
// the 9 correlations; C=128 reduced in 32 chained K=4 WMMA steps.
// ---------------------------------------------------------------------------

typedef __attribute__((ext_vector_type(2))) float v2f;
typedef __attribute__((ext_vector_type(8))) float v8f;

#define RR      4
#define NDISP   9
#define CCH     128
#define HHH     128
#define WWW     256
#define TILE_P  16
#define TILE_Q  32
#define APAD    132         // f32 row stride of [pos][c] tiles (bank-spread)
#define DPAD    17          // f32 row stride of D scratch (overlaid on wT)
#define NWAVES  3
#define BLOCKT  (NWAVES * 32)

#if __has_builtin(__builtin_amdgcn_global_load_async_to_lds_b32)
#define HAVE_ASYNC_LDS 1
typedef __attribute__((address_space(1))) int g_i32;   // global int
typedef __attribute__((address_space(3))) int l_i32;   // LDS int
#else
#define HAVE_ASYNC_LDS 0
#endif

__device__ __forceinline__ void async_wait0() {
#if __has_builtin(__builtin_amdgcn_s_wait_asynccnt)
  __builtin_amdgcn_s_wait_asynccnt(0);
#elif HAVE_ASYNC_LDS
  asm volatile("s_wait_asynccnt 0" ::: "memory");
#endif
}

// one 4-byte global -> LDS copy (async when available)
__device__ __forceinline__ void g2lds_f32(float* ldst, const float* gsrc) {
#if HAVE_ASYNC_LDS
  __builtin_amdgcn_global_load_async_to_lds_b32(
      (g_i32*)gsrc, (l_i32*)ldst, /*offset=*/0, /*cpol=*/0);
#else
  *ldst = *gsrc;
#endif
}

__global__ __launch_bounds__(BLOCKT)
void cost_volume_wmma(const float* __restrict__ x,
                      const float* __restrict__ warped,
                      float* __restrict__ out)
{
  __shared__ float xT[TILE_P * APAD];               // x tile,   [p][c] f32
  __shared__ float wT[NWAVES][TILE_Q * APAD];       // warped,   [q][c] f32 (per wave)

  const int tid  = threadIdx.x;
  const int lane = tid & 31;
  const int wave = tid >> 5;
  const int p0   = blockIdx.x * TILE_P;
  const int h    = blockIdx.y;
  const int b    = blockIdx.z;

  // ---- stage x tile: global [c][w] -> LDS [p][c] (transpose in flight) ----
  for (int e = tid; e < CCH * TILE_P; e += BLOCKT) {
    const int c = e >> 4;
    const int p = e & 15;
    g2lds_f32(&xT[p * APAD + c],
              &x[(((size_t)b * CCH + c) * HHH + h) * WWW + p0 + p]);
  }
  async_wait0();
  __syncthreads();

  const float scale = 1.0f / ((float)CCH * (float)(NDISP * NDISP));

  float* wbuf = &wT[wave][0];
  float* dbuf = wbuf;                    // D-extraction scratch overlays wbuf

  // per-lane fragment addressing: 32-bit A/B layout, K pairs split by half-wave
  const int    frow = lane & 15;         // A: M row / B: N col
  const int    kh   = (lane < 16) ? 0 : 2;
  const float* aB   = &xT[frow * APAD + kh];
  const float* b0B  = &wbuf[frow * APAD + kh];          // q-tile 0: q = 0..15
  const float* b1B  = &wbuf[(frow + 16) * APAD + kh];   // q-tile 1: q = 16..31

  for (int diIdx = wave; diIdx < NDISP; diIdx += NWAVES) {
    // ---- stage warped tile: row h+di, cols p0-4 .. p0+27, zero-padded -----
    const int  hh    = h + diIdx - RR;
    const int  qAbs  = p0 - RR + lane;
    const bool valid = (hh >= 0) & (hh < HHH) & (qAbs >= 0) & (qAbs < WWW);
    if (valid) {
      const float* src = warped + ((size_t)b * CCH * HHH + hh) * WWW + qAbs;
      #pragma unroll 8
      for (int c = 0; c < CCH; ++c)
        g2lds_f32(&wbuf[lane * APAD + c], src + (size_t)c * HHH * WWW);
    } else {
      const v2f z = {0.0f, 0.0f};
      #pragma unroll 8
      for (int c = 0; c < CCH; c += 2)
        *(v2f*)&wbuf[lane * APAD + c] = z;
    }
    async_wait0();
    __syncthreads();                     // cross-lane LDS visibility

    // ---- K-reduction: 32 chained v_wmma_f32_16x16x4_f32 per accumulator ---
    v8f acc0 = {};
    v8f acc1 = {};
    #pragma unroll
    for (int k0 = 0; k0 < CCH; k0 += 4) {
      v2f a  = *(const v2f*)(aB  + k0);
      v2f b0 = *(const v2f*)(b0B + k0);
      acc0 = __builtin_amdgcn_wmma_f32_16x16x4_f32(false, a, false, b0,
                                                   (short)0, acc0, false, false);
      v2f b1 = *(const v2f*)(b1B + k0);
      acc1 = __builtin_amdgcn_wmma_f32_16x16x4_f32(false, a, false, b1,
                                                   (short)0, acc1, false, false);
    }

    // ---- bounce D tiles through LDS to extract the dj band ----------------
    {
      const int ncol = lane & 15;
      const int mrow = (lane < 16) ? 0 : 8;
      #pragma unroll
      for (int r = 0; r < 8; ++r) {
        dbuf[ ncol       * DPAD + mrow + r] = acc0[r];
        dbuf[(ncol + 16) * DPAD + mrow + r] = acc1[r];
      }
    }
    __syncthreads();                     // cross-lane LDS visibility

    // ---- write 9 dj x 16 w outputs for this (b, di, h, strip) -------------
    const int dBase = diIdx * NDISP;     // (di+R)*9
    for (int e = lane; e < NDISP * TILE_P; e += 32) {
      const int djIdx = e >> 4;          // 0..8
      const int i     = e & 15;          // w offset in strip
      const int q     = i + djIdx;       // D scratch row (0..23)
      const float val = dbuf[q * DPAD + i] * scale;
      out[(((size_t)b * (NDISP * NDISP) + dBase + djIdx) * HHH + h) * WWW + p0 + i] = val;
    }
    __syncthreads();                     // wbuf/dbuf reuse next di
  }
}

extern "C" void kernel_launch(void* const* d_in, const int* in_sizes, int n_in,
                              void* d_out, int out_size, void* d_ws, size_t ws_size,
                              hipStream_t stream) {
  (void)in_sizes; (void)n_in; (void)d_ws; (void)ws_size; (void)out_size;
  const float* x      = (const float*)d_in[0];
  const float* warped = (const float*)d_in[1];
  float*       out    = (float*)d_out;

  dim3 grid(WWW / TILE_P, HHH, 8);   // (16, 128, 8) = 16384 blocks
  dim3 block(BLOCKT);                // 3 wave32
  cost_volume_wmma<<<grid, block, 0, stream>>>(x, warped, out);
}